// IAL_31018253811874
// MI455X (gfx1250) — compile-verified
//
#include <hip/hip_runtime.h>
#include <hip/hip_bf16.h>
#include <math.h>

#define N_ROWS 4096
#define DIM    512
#define INV_T  2.0f      // 1 / TEMPERATURE, TEMPERATURE = 0.5
#define EPS    1e-8f

typedef __attribute__((ext_vector_type(2))) float v2f;
typedef __attribute__((ext_vector_type(8))) float v8f;

// ---------------------------------------------------------------------------
// Kernel 1: row-normalize z and z_adv, compute expadv[i] = exp(2*cos(z_i,za_i))
// One block (256 threads, 8 waves) per row; each thread owns 2 elements.
// ---------------------------------------------------------------------------
__global__ __launch_bounds__(256) void k_normalize(
    const float* __restrict__ z, const float* __restrict__ za,
    float* __restrict__ zn, float* __restrict__ expadv)
{
    const int row  = blockIdx.x;
    const int tid  = threadIdx.x;
    const int wave = tid >> 5;
    const int lane = tid & 31;
    const size_t base = (size_t)row * DIM;

    float x0 = z[base + tid];
    float x1 = z[base + tid + 256];
    float y0 = za[base + tid];
    float y1 = za[base + tid + 256];

    float szz = fmaf(x0, x0, x1 * x1);   // |z|^2 partial
    float saa = fmaf(y0, y0, y1 * y1);   // |za|^2 partial
    float sza = fmaf(x0, y0, x1 * y1);   // z.za partial

    // wave32 reduction
    #pragma unroll
    for (int off = 16; off > 0; off >>= 1) {
        szz += __shfl_xor(szz, off, 32);
        saa += __shfl_xor(saa, off, 32);
        sza += __shfl_xor(sza, off, 32);
    }

    __shared__ float r0[8], r1[8], r2[8];
    if (lane == 0) { r0[wave] = szz; r1[wave] = saa; r2[wave] = sza; }
    __syncthreads();

    float t0 = 0.f, t1 = 0.f, t2 = 0.f;
    #pragma unroll
    for (int w = 0; w < 8; ++w) { t0 += r0[w]; t1 += r1[w]; t2 += r2[w]; }

    // torch CosineSimilarity: divide by max(norm, eps)
    const float inv_nz = 1.0f / fmaxf(sqrtf(t0), EPS);
    const float inv_na = 1.0f / fmaxf(sqrtf(t1), EPS);

    zn[base + tid]       = x0 * inv_nz;
    zn[base + tid + 256] = x1 * inv_nz;

    if (tid == 0)
        expadv[row] = __expf(t2 * inv_nz * inv_na * INV_T);
}

// ---------------------------------------------------------------------------
// Kernel 2: masked Gram-matrix row sums via V_WMMA_F32_16X16X4_F32.
// Block = 16-row tile I (staged in LDS); 8 waves each sweep 32 column tiles J.
// intra[i] = sum_j exp(2*zn_i.zn_j) over same-label, off-diagonal j.
// Block then adds sum_i log1p(intra[i]/expadv[i]) / N into *out.
// ---------------------------------------------------------------------------
__global__ __launch_bounds__(256) void k_gram_loss(
    const float* __restrict__ zn, const float* __restrict__ expadv,
    const int* __restrict__ labels, float* __restrict__ out)
{
    __shared__ __align__(16) float sA[16 * DIM];   // 32 KB A tile
    __shared__ float s_intra[16];
    __shared__ int   s_lab[16];

    const int I    = blockIdx.x;        // row-tile index 0..255
    const int tid  = threadIdx.x;
    const int wave = tid >> 5;
    const int lane = tid & 31;
    const int halfsel = lane >> 4;      // 0: K pair {0,1}, 1: K pair {2,3}
    const int lrow    = lane & 15;      // M (for A) / N (for B,C,D)

    // Stage the 16x512 f32 row tile into LDS (float4 vectorized).
    const float* zrowI = zn + (size_t)I * 16 * DIM;
    for (int t = tid; t < 16 * DIM / 4; t += 256)
        ((float4*)sA)[t] = ((const float4*)zrowI)[t];
    if (tid < 16) { s_intra[tid] = 0.0f; s_lab[tid] = labels[I * 16 + tid]; }
    __syncthreads();

    const float* arow = sA + lrow * DIM + 2 * halfsel;

    float acc[8];
    #pragma unroll
    for (int r = 0; r < 8; ++r) acc[r] = 0.0f;

    for (int J = wave; J < N_ROWS / 16; J += 8) {
        const int jcol = J * 16 + lrow;                    // this lane's N column
        const float* brow = zn + (size_t)jcol * DIM + 2 * halfsel;

        v8f c = {};
        #pragma unroll 4
        for (int k = 0; k < DIM; k += 4) {
            v2f a = *(const v2f*)(arow + k);   // A: lane -> (M=lrow, K=k+2*halfsel..)
            v2f b = *(const v2f*)(brow + k);   // B: lane -> (N=lrow, K=k+2*halfsel..)
            c = __builtin_amdgcn_wmma_f32_16x16x4_f32(
                    /*neg_a=*/false, a, /*neg_b=*/false, b,
                    /*c_mod=*/(short)0, c, /*reuse_a=*/false, /*reuse_b=*/false);
        }

        // Masked exp epilogue. C/D layout: vgpr r, lanes 0-15 -> M=r,
        // lanes 16-31 -> M=r+8; N = lane&15 in both halves.
        const int lab_j = labels[jcol];
        #pragma unroll
        for (int r = 0; r < 8; ++r) {
            const int m    = r + 8 * halfsel;
            const int irow = I * 16 + m;
            const bool keep = (s_lab[m] == lab_j) && (irow != jcol);
            acc[r] += keep ? __expf(c[r] * INV_T) : 0.0f;
        }
    }

    // Reduce each acc[r] across the 16 lanes of its half-wave (sum over N).
    #pragma unroll
    for (int r = 0; r < 8; ++r) {
        float v = acc[r];
        #pragma unroll
        for (int off = 8; off > 0; off >>= 1)
            v += __shfl_xor(v, off, 32);       // off<16: stays within half
        acc[r] = v;
    }
    if (lrow == 0) {                            // lanes 0 and 16 of each wave
        #pragma unroll
        for (int r = 0; r < 8; ++r)
            atomicAdd(&s_intra[r + 8 * halfsel], acc[r]);   // ds_add_f32
    }
    __syncthreads();

    // Per-row loss for the 16 rows of tile I, then one global atomic per block.
    if (tid < 16) {
        const float e    = expadv[I * 16 + tid];
        float loss = log1pf(s_intra[tid] / e);  // -log(e/(e+intra))
        #pragma unroll
        for (int off = 8; off > 0; off >>= 1)
            loss += __shfl_xor(loss, off, 32);
        if (tid == 0)
            atomicAdd(out, loss * (1.0f / (float)N_ROWS));
    }
}

// ---------------------------------------------------------------------------
extern "C" void kernel_launch(void* const* d_in, const int* in_sizes, int n_in,
                              void* d_out, int out_size, void* d_ws, size_t ws_size,
                              hipStream_t stream)
{
    const float* z      = (const float*)d_in[0];   // [4096,512] f32
    const float* za     = (const float*)d_in[1];   // [4096,512] f32
    const int*   labels = (const int*)d_in[2];     // [4096] int
    float*       out    = (float*)d_out;           // scalar f32

    float* zn     = (float*)d_ws;                  // 4096*512 f32 = 8 MB
    float* expadv = zn + (size_t)N_ROWS * DIM;     // 4096 f32

    hipMemsetAsync(out, 0, sizeof(float), stream); // harness poisons d_out

    k_normalize<<<N_ROWS, 256, 0, stream>>>(z, za, zn, expadv);
    k_gram_loss<<<N_ROWS / 16, 256, 0, stream>>>(zn, expadv, labels, out);
}